// LinearEncoder_89335319757132
// MI455X (gfx1250) — compile-verified
//
#include <hip/hip_runtime.h>

typedef float v2f __attribute__((ext_vector_type(2)));
typedef float v8f __attribute__((ext_vector_type(8)));

#define D 64

// ---------------- degree: deg[i] = 1 (self loop) + #edges targeting i ----------------
__global__ void k_deg_init(float* __restrict__ deg, int n) {
  int i = blockIdx.x * blockDim.x + threadIdx.x;
  if (i < n) deg[i] = 1.0f;  // self-loop contributes 1 to every node
}

__global__ void k_deg_accum(const long long* __restrict__ col, float* __restrict__ deg, int e) {
  int i = blockIdx.x * blockDim.x + threadIdx.x;
  if (i < e) unsafeAtomicAdd(&deg[(int)col[i]], 1.0f);  // global_atomic_add_f32
}

__global__ void k_deg_rsqrt(float* __restrict__ deg, int n) {
  int i = blockIdx.x * blockDim.x + threadIdx.x;
  if (i < n) {
    float d = deg[i];
    deg[i] = d > 0.0f ? rsqrtf(d) : 0.0f;  // in place: deg -> deg^{-1/2}
  }
}

// ---------------- xw = x @ W via V_WMMA_F32_16X16X4_F32 ----------------
// One wave computes a 16-row strip of all 64 output columns (4 N-tiles).
// A-frag (16x4 f32): lane = half*16 + m; VGPR0/1 hold K = 2*half, 2*half+1.
// B-frag (4x16 f32): lane = half*16 + ncol; VGPR0/1 hold K rows 2*half, 2*half+1.
// C/D (16x16 f32): VGPR r, lane half*16 + ncol -> row m0 + half*8 + r.
__global__ void k_gemm_wmma(const float* __restrict__ x, const float* __restrict__ W,
                            float* __restrict__ xw, int n, int nTiles) {
  const int wave = (int)((blockIdx.x * (unsigned)blockDim.x + threadIdx.x) >> 5);
  if (wave >= nTiles) return;            // wave-uniform guard: EXEC stays all-ones
  const int lane = threadIdx.x & 31;
  const int half = lane >> 4;
  const int l16  = lane & 15;
  const int m0   = wave * 16;

  const int arow = min(m0 + l16, n - 1); // clamp (N divisible by 16 here, but be safe)
  const float* __restrict__ ap = x + (size_t)arow * D;

  v8f acc0 = {}, acc1 = {}, acc2 = {}, acc3 = {};
#pragma unroll
  for (int k = 0; k < D; k += 4) {
    v2f a;
    a.x = ap[k + 2 * half + 0];
    a.y = ap[k + 2 * half + 1];
    const float* __restrict__ w0 = W + (size_t)(k + 2 * half) * D + l16;  // K row 2*half
    const float* __restrict__ w1 = w0 + D;                                // K row 2*half+1
    v2f b0, b1, b2, b3;
    b0.x = w0[0];  b0.y = w1[0];
    b1.x = w0[16]; b1.y = w1[16];
    b2.x = w0[32]; b2.y = w1[32];
    b3.x = w0[48]; b3.y = w1[48];
    acc0 = __builtin_amdgcn_wmma_f32_16x16x4_f32(false, a, false, b0, (short)0, acc0, false, false);
    acc1 = __builtin_amdgcn_wmma_f32_16x16x4_f32(false, a, false, b1, (short)0, acc1, false, false);
    acc2 = __builtin_amdgcn_wmma_f32_16x16x4_f32(false, a, false, b2, (short)0, acc2, false, false);
    acc3 = __builtin_amdgcn_wmma_f32_16x16x4_f32(false, a, false, b3, (short)0, acc3, false, false);
  }

#pragma unroll
  for (int r = 0; r < 8; ++r) {
    int m = m0 + half * 8 + r;
    if (m < n) {
      float* __restrict__ o = xw + (size_t)m * D + l16;
      o[0]  = acc0[r];
      o[16] = acc1[r];
      o[32] = acc2[r];
      o[48] = acc3[r];
    }
  }
}

// ---------------- out = b + dinv[i]^2 * xw[i]  (self-loop term + bias, full init) ----------------
__global__ void k_out_init(const float* __restrict__ xw, const float* __restrict__ dinv,
                           const float* __restrict__ bias, float* __restrict__ out, int n) {
  int tid = blockIdx.x * blockDim.x + threadIdx.x;
  if (tid >= n * 16) return;
  int i = tid >> 4;
  int d = (tid & 15) * 4;
  float s = dinv[i];
  s *= s;
  const float4 v  = *(const float4*)(xw + (size_t)i * D + d);
  const float4 bb = *(const float4*)(bias + d);
  float4 o;
  o.x = bb.x + s * v.x;
  o.y = bb.y + s * v.y;
  o.z = bb.z + s * v.z;
  o.w = bb.w + s * v.w;
  *(float4*)(out + (size_t)i * D + d) = o;
}

// ---------------- per-edge gather / normalize / scatter-add ----------------
// 16 threads per edge, float4 per thread; 4 fp32 global atomics per thread.
// Working set (~77 MB) fits the 192 MB L2, so this runs at L2 bandwidth.
__global__ void k_scatter(const long long* __restrict__ row, const long long* __restrict__ col,
                          const float* __restrict__ dinv, const float* __restrict__ xw,
                          float* __restrict__ out, int e) {
  long long tid = (long long)blockIdx.x * blockDim.x + threadIdx.x;
  if (tid >= (long long)e * 16) return;
  int idx = (int)(tid >> 4);
  int d   = ((int)tid & 15) * 4;
  int r = (int)row[idx];
  int c = (int)col[idx];
  float w = dinv[r] * dinv[c];
  const float4 v = *(const float4*)(xw + (size_t)r * D + d);
  float* __restrict__ o = out + (size_t)c * D + d;
  unsafeAtomicAdd(o + 0, w * v.x);
  unsafeAtomicAdd(o + 1, w * v.y);
  unsafeAtomicAdd(o + 2, w * v.z);
  unsafeAtomicAdd(o + 3, w * v.w);
}

extern "C" void kernel_launch(void* const* d_in, const int* in_sizes, int n_in,
                              void* d_out, int out_size, void* d_ws, size_t ws_size,
                              hipStream_t stream) {
  const float*     x  = (const float*)d_in[0];      // [N, 64]
  const long long* ei = (const long long*)d_in[1];  // [2, E] int64
  const float*     W  = (const float*)d_in[2];      // [64, 64]
  const float*     b  = (const float*)d_in[3];      // [64]
  float* out = (float*)d_out;                       // [N, 64]

  const int n = in_sizes[0] / D;
  const int e = in_sizes[1] / 2;
  const long long* row = ei;      // sources
  const long long* col = ei + e;  // targets

  // workspace: xw [n*64] floats, then dinv [n] floats (deg reused in place)
  float* xw   = (float*)d_ws;
  float* dinv = xw + (size_t)n * D;

  const int B = 256;

  k_deg_init <<<(n + B - 1) / B, B, 0, stream>>>(dinv, n);
  k_deg_accum<<<(e + B - 1) / B, B, 0, stream>>>(col, dinv, e);
  k_deg_rsqrt<<<(n + B - 1) / B, B, 0, stream>>>(dinv, n);

  const int nTiles = (n + 15) / 16;           // 6250 waves for N=100000
  const int wavesPerBlock = B / 32;           // wave32
  k_gemm_wmma<<<(nTiles + wavesPerBlock - 1) / wavesPerBlock, B, 0, stream>>>(x, W, xw, n, nTiles);

  const int initThreads = n * 16;
  k_out_init<<<(initThreads + B - 1) / B, B, 0, stream>>>(xw, dinv, b, out, n);

  const long long scatThreads = (long long)e * 16;
  k_scatter<<<(int)((scatThreads + B - 1) / B), B, 0, stream>>>(row, col, dinv, xw, out, e);
}